// GCN_188978561632
// MI455X (gfx1250) — compile-verified
//
#include <hip/hip_runtime.h>
#include <math.h>

// ---------------------------------------------------------------------------
// GCN pipeline for MI455X (gfx1250, wave32).
// FP32 throughout (working set is L2-resident at 192MB; precision is free).
// GEMMs: V_WMMA_F32_16X16X4_F32, one wave per 16-row stripe holding all
// Fo/16 output tiles in registers (A read once). Edge phase: float4 gathers
// + hardware global_atomic_add_f32 scatter.
// ---------------------------------------------------------------------------

typedef __attribute__((ext_vector_type(2))) float v2f;
typedef __attribute__((ext_vector_type(8))) float v8f;

__device__ __forceinline__ void atomAddF(float* p, float v) {
  unsafeAtomicAdd(p, v);  // hardware global_atomic_add_f32
}

// -------------------------- utility kernels --------------------------------

__global__ void k_fill(float* __restrict__ p, float val, size_t n) {
  size_t i = (size_t)blockIdx.x * blockDim.x + threadIdx.x;
  if (i < n) p[i] = val;
}

// deg[dst] += 1 for every real edge (deg pre-initialized to 1.0 = self loop)
__global__ void k_deg(const int* __restrict__ dst, float* __restrict__ deg, int E) {
  int e = blockIdx.x * blockDim.x + threadIdx.x;
  if (e >= E) return;
  if (e + 4096 < E) __builtin_prefetch(&dst[e + 4096], 0, 1);
  atomAddF(&deg[dst[e]], 1.0f);
}

// deg -> D^{-1/2} in place
__global__ void k_dinv(float* __restrict__ d, int N) {
  int i = blockIdx.x * blockDim.x + threadIdx.x;
  if (i >= N) return;
  float v = d[i];
  d[i] = (v > 0.0f) ? rsqrtf(v) : 0.0f;
}

// norm[e] = dinv[src]*dinv[dst]
__global__ void k_norm(const int* __restrict__ src, const int* __restrict__ dst,
                       const float* __restrict__ dinv, float* __restrict__ nrm, int E) {
  int e = blockIdx.x * blockDim.x + threadIdx.x;
  if (e >= E) return;
  nrm[e] = dinv[src[e]] * dinv[dst[e]];
}

// --------------------------- WMMA fp32 GEMM --------------------------------
// O[N, TN*16] = A[N,Fi] @ W[Fi, TN*16]
// One wave per 16-row stripe; TN accumulator tiles live in registers so the
// A stripe is read from memory exactly once.
// A layout : lanes 0-15 M=0..15 (K=k0,k0+1), lanes 16-31 M=0..15 (K=k0+2,k0+3)
// B layout : lanes 0-15 N=0..15 (K=k0,k0+1), lanes 16-31 N=0..15 (K=k0+2,k0+3)
// C/D      : VGPR j, lanes 0-15 -> M=j, lanes 16-31 -> M=j+8
template <int TN>
__global__ void k_gemm_wmma(const float* __restrict__ A, const float* __restrict__ W,
                            float* __restrict__ O, int N, int Fi) {
  const int Fo = TN * 16;
  unsigned wave   = (unsigned)(blockIdx.x * blockDim.x + threadIdx.x) >> 5;
  unsigned lane   = threadIdx.x & 31u;
  unsigned tilesM = ((unsigned)N + 15u) >> 4;
  if (wave >= tilesM) return;  // wave-uniform exit: EXEC all-ones for WMMA

  unsigned l15 = lane & 15u;
  unsigned hi  = lane >> 4;   // 0 | 1
  unsigned kh  = hi << 1;     // 0 | 2
  unsigned row = wave * 16u + l15;
  if (row >= (unsigned)N) row = (unsigned)N - 1u;

  v8f c[TN];
#pragma unroll
  for (int t = 0; t < TN; ++t) c[t] = {};

  for (int k0 = 0; k0 < Fi; k0 += 4) {
    v2f a;
    const float* ap = A + (size_t)row * Fi + k0 + kh;
    a.x = ap[0];
    a.y = ap[1];
    const float* bp = W + (size_t)(k0 + kh) * Fo + l15;
#pragma unroll
    for (int t = 0; t < TN; ++t) {
      v2f b;
      b.x = bp[t * 16];
      b.y = bp[t * 16 + Fo];
      c[t] = __builtin_amdgcn_wmma_f32_16x16x4_f32(false, a, false, b,
                                                   (short)0, c[t], false, false);
    }
  }

  unsigned m0 = wave * 16u + (hi << 3);
#pragma unroll
  for (int t = 0; t < TN; ++t) {
    unsigned col = t * 16u + l15;
#pragma unroll
    for (int j = 0; j < 8; ++j) {
      unsigned m = m0 + (unsigned)j;
      if (m < (unsigned)N) O[(size_t)m * Fo + col] = c[t][j];
    }
  }
}

// -------------------- accumulator init: self-loop + bias -------------------
// acc[i,f] = dinv[i]^2 * hw[i,f] + b[f]   (replaces zero-fill before scatter)
__global__ void k_init_acc(float* __restrict__ acc, const float* __restrict__ hw,
                           const float* __restrict__ dinv, const float* __restrict__ bias,
                           int N, int foShift) {
  size_t tid = (size_t)blockIdx.x * blockDim.x + threadIdx.x;
  size_t total = (size_t)N << foShift;
  if (tid >= total) return;
  unsigned i = (unsigned)(tid >> foShift);
  unsigned f = (unsigned)(tid & (((size_t)1 << foShift) - 1));
  float di = dinv[i];
  acc[tid] = di * di * hw[tid] + bias[f];
}

// ------------------------------ edge scatter -------------------------------
// acc[dst[e], :] += norm[e] * h[src[e], :]
// One thread per (edge, 4-feature chunk); consecutive threads cover one row.
__global__ void k_scatter(const float* __restrict__ h, const int* __restrict__ src,
                          const int* __restrict__ dst, const float* __restrict__ nrm,
                          float* __restrict__ acc, int E, int cShift, int foShift) {
  unsigned tid = (unsigned)(blockIdx.x * blockDim.x + threadIdx.x);
  unsigned e = tid >> cShift;
  unsigned c = tid & ((1u << cShift) - 1u);
  if (e >= (unsigned)E) return;
  if (c == 0 && e + 2048u < (unsigned)E) {
    __builtin_prefetch(&src[e + 2048u], 0, 1);
    __builtin_prefetch(&dst[e + 2048u], 0, 1);
    __builtin_prefetch(&nrm[e + 2048u], 0, 1);
  }
  int s = src[e];
  int d = dst[e];
  float w = nrm[e];
  const float4 v = *reinterpret_cast<const float4*>(h + ((size_t)s << foShift) + (c << 2));
  float* p = acc + ((size_t)d << foShift) + (c << 2);
  atomAddF(p + 0, v.x * w);
  atomAddF(p + 1, v.y * w);
  atomAddF(p + 2, v.z * w);
  atomAddF(p + 3, v.w * w);
}

// ------------------------------ in-place tanh ------------------------------
__global__ void k_tanh(float* __restrict__ acc, size_t total) {
  size_t tid = (size_t)blockIdx.x * blockDim.x + threadIdx.x;
  if (tid >= total) return;
  acc[tid] = tanhf(acc[tid]);
}

// ------------------------------ mean pool ----------------------------------
__global__ void k_pool(const float* __restrict__ h, const int* __restrict__ batch,
                       float* __restrict__ sums, float* __restrict__ cnt, int N) {
  unsigned tid = (unsigned)(blockIdx.x * blockDim.x + threadIdx.x);
  unsigned i = tid >> 4;  // 16 chunks of 4 feats = 64 feats
  unsigned c = tid & 15u;
  if (i >= (unsigned)N) return;
  int g = batch[i];
  const float4 v = *reinterpret_cast<const float4*>(h + (size_t)i * 64 + (c << 2));
  float* p = sums + (size_t)g * 64 + (c << 2);
  atomAddF(p + 0, v.x);
  atomAddF(p + 1, v.y);
  atomAddF(p + 2, v.z);
  atomAddF(p + 3, v.w);
  if (c == 0) atomAddF(&cnt[g], 1.0f);
}

__global__ void k_head(const float* __restrict__ sums, const float* __restrict__ cnt,
                       const float* __restrict__ Wc, const float* __restrict__ bc,
                       float* __restrict__ out, int G) {
  int g = blockIdx.x * blockDim.x + threadIdx.x;
  if (g >= G) return;
  float inv = 1.0f / fmaxf(cnt[g], 1.0f);
  float acc = 0.0f;
#pragma unroll 8
  for (int f = 0; f < 64; ++f) acc = fmaf(sums[g * 64 + f] * inv, Wc[f], acc);
  out[g] = acc + bc[0];
}

// ------------------------------- launcher ----------------------------------

extern "C" void kernel_launch(void* const* d_in, const int* in_sizes, int n_in,
                              void* d_out, int out_size, void* d_ws, size_t ws_size,
                              hipStream_t stream) {
  (void)n_in; (void)ws_size;
  const float* x     = (const float*)d_in[0];
  const int*   ei    = (const int*)d_in[1];
  const int*   batch = (const int*)d_in[2];
  const float* Wl[5] = {(const float*)d_in[3], (const float*)d_in[5], (const float*)d_in[7],
                        (const float*)d_in[9], (const float*)d_in[11]};
  const float* bl[5] = {(const float*)d_in[4], (const float*)d_in[6], (const float*)d_in[8],
                        (const float*)d_in[10], (const float*)d_in[12]};
  const float* Wc = (const float*)d_in[13];
  const float* bc = (const float*)d_in[14];

  const int N = in_sizes[2];      // 100000
  const int E = in_sizes[1] / 2;  // 1600000
  const int G = out_size;         // 128
  const int* src = ei;
  const int* dst = ei + E;

  // workspace carve-up (256B aligned)
  char* ws = (char*)d_ws;
  const size_t SB = (size_t)N * 128 * sizeof(float);
  size_t off = 0;
  auto al = [](size_t v) { return (v + 255) & ~(size_t)255; };
  float* buf[3];
  for (int i = 0; i < 3; ++i) { buf[i] = (float*)(ws + off); off = al(off + SB); }
  float* nrm  = (float*)(ws + off); off = al(off + (size_t)E * sizeof(float));
  float* dinv = (float*)(ws + off); off = al(off + (size_t)N * sizeof(float));
  float* sums = (float*)(ws + off); off = al(off + (size_t)G * 64 * sizeof(float));
  float* cnt  = (float*)(ws + off); off = al(off + (size_t)G * sizeof(float));

  const int B = 256;  // 8 waves/block on wave32
  auto blocks = [](size_t n, int b) { return (unsigned)((n + (size_t)b - 1) / (size_t)b); };

  // normalization: deg (incl. self loop) -> D^{-1/2} -> per-edge norm
  k_fill<<<blocks(N, B), B, 0, stream>>>(dinv, 1.0f, (size_t)N);
  k_deg<<<blocks(E, B), B, 0, stream>>>(dst, dinv, E);
  k_dinv<<<blocks(N, B), B, 0, stream>>>(dinv, N);
  k_norm<<<blocks(E, B), B, 0, stream>>>(src, dst, dinv, nrm, E);
  k_fill<<<blocks((size_t)G * 64, B), B, 0, stream>>>(sums, 0.0f, (size_t)G * 64);
  k_fill<<<blocks(G, B), B, 0, stream>>>(cnt, 0.0f, (size_t)G);

  const int FiA[5]    = {64, 16, 128, 128, 64};
  const int FoA[5]    = {16, 128, 128, 64, 64};
  const int rotIn[5]  = {-1, 1, 0, 2, 1};
  const int rotHw[5]  = { 0, 2, 1, 0, 2};
  const int rotAcc[5] = { 1, 0, 2, 1, 0};

  const unsigned gemmThreads = (unsigned)((N + 15) / 16) * 32u;  // one wave / 16 rows

  for (int l = 0; l < 5; ++l) {
    const float* hin = (l == 0) ? x : buf[rotIn[l]];
    float* hw  = buf[rotHw[l]];
    float* acc = buf[rotAcc[l]];
    const int Fi = FiA[l], Fo = FoA[l];

    if (Fo == 16) {
      k_gemm_wmma<1><<<blocks(gemmThreads, B), B, 0, stream>>>(hin, Wl[l], hw, N, Fi);
    } else if (Fo == 64) {
      k_gemm_wmma<4><<<blocks(gemmThreads, B), B, 0, stream>>>(hin, Wl[l], hw, N, Fi);
    } else {
      k_gemm_wmma<8><<<blocks(gemmThreads, B), B, 0, stream>>>(hin, Wl[l], hw, N, Fi);
    }

    const int foShift = (Fo == 128) ? 7 : (Fo == 64) ? 6 : 4;  // log2(Fo)
    k_init_acc<<<blocks((size_t)N << foShift, B), B, 0, stream>>>(acc, hw, dinv, bl[l],
                                                                  N, foShift);

    const int cShift = foShift - 2;  // log2(Fo/4)
    k_scatter<<<blocks((size_t)E << cShift, B), B, 0, stream>>>(hw, src, dst, nrm, acc,
                                                                E, cShift, foShift);

    k_tanh<<<blocks((size_t)N << foShift, B), B, 0, stream>>>(acc, (size_t)N << foShift);
  }

  // final node features live in buf[0]
  k_pool<<<blocks((size_t)N * 16, B), B, 0, stream>>>(buf[0], batch, sums, cnt, N);
  k_head<<<blocks(G, 128), 128, 0, stream>>>(sums, cnt, Wc, bc, (float*)d_out, G);
}